// model_1460288881248
// MI455X (gfx1250) — compile-verified
//
#include <hip/hip_runtime.h>
#include <math.h>

typedef float v2f __attribute__((ext_vector_type(2)));
typedef float v8f __attribute__((ext_vector_type(8)));

#define HD 32
#define PD 8

__device__ __forceinline__ v8f wmma4(v2f a, v2f b, v8f c) {
  // D = A(16x4 f32) * B(4x16 f32) + C(16x16 f32), wave32
  return __builtin_amdgcn_wmma_f32_16x16x4_f32(false, a, false, b, (short)0, c,
                                               false, false);
}

__device__ __forceinline__ v8f splat8(float v) {
  v8f r;
#pragma unroll
  for (int i = 0; i < 8; ++i) r[i] = v;
  return r;
}

// accurate-enough branch-free reciprocal: v_rcp_f32 + one Newton step
__device__ __forceinline__ float rcp1(float x) {
  float r = __builtin_amdgcn_rcpf(x);
  return r * (2.0f - x * r);
}

// clamp so exp never returns inf/0-denorm (keeps rcp/Newton NaN-free);
// tanh/sigmoid are saturated to f32 precision well before |arg|=30
__device__ __forceinline__ float exp_clamped(float x) {
  return __expf(fminf(fmaxf(x, -30.0f), 30.0f));
}

// ---------------- zero scratch ----------------
__global__ void zero_kernel(float* __restrict__ p, long long n) {
  long long i = (long long)blockIdx.x * blockDim.x + threadIdx.x;
  long long stride = (long long)gridDim.x * blockDim.x;
  for (; i < n; i += stride) p[i] = 0.0f;
}

// ---------------- degree via atomics ----------------
__global__ void deg_kernel(const int* __restrict__ dst, float* __restrict__ deg, int E) {
  int e = blockIdx.x * blockDim.x + threadIdx.x;
  if (e < E) atomicAdd(&deg[dst[e]], 1.0f);
}

// ---------------- dinv = (deg+1)^-0.5 ----------------
__global__ void dinv_kernel(const float* __restrict__ deg, float* __restrict__ dinv, int N) {
  int n = blockIdx.x * blockDim.x + threadIdx.x;
  if (n < N) {
    float d = deg[n] + 1.0f;
    float y = rsqrtf(d);
    y = y * (1.5f - 0.5f * d * y * y);  // Newton refine
    dinv[n] = y;
  }
}

// ---- fuse gate weights: WL = W @ Ltop (4x32), bL = b @ Ltop + lb (32) ----
// u_t = (xa_t @ W + b) @ Ltop + lb  ==  xa_t @ WL + bL   (H0 == 0)
__global__ void fuse_kernel(const float* __restrict__ Wz, const float* __restrict__ bz,
                            const float* __restrict__ LzW, const float* __restrict__ Lzb,
                            const float* __restrict__ Wh, const float* __restrict__ bh,
                            const float* __restrict__ LhW, const float* __restrict__ Lhb,
                            float* __restrict__ WzL, float* __restrict__ bzL,
                            float* __restrict__ WhL, float* __restrict__ bhL) {
  int t = threadIdx.x;  // 0..319 : two gates x (128 weight entries + 32 bias entries)
  if (t >= 320) return;
  int gate = t / 160;
  int r = t % 160;
  const float* W = gate ? Wh : Wz;
  const float* b = gate ? bh : bz;
  const float* L = gate ? LhW : LzW;  // use rows 0..31 (top half) only
  const float* lb = gate ? Lhb : Lzb;
  float* WL = gate ? WhL : WzL;
  float* bL = gate ? bhL : bzL;
  if (r < 128) {
    int k = r >> 5, col = r & 31;
    float s = 0.0f;
#pragma unroll
    for (int j = 0; j < HD; ++j) s += W[k * HD + j] * L[j * HD + col];
    WL[k * HD + col] = s;
  } else {
    int col = r - 128;
    float s = lb[col];
#pragma unroll
    for (int j = 0; j < HD; ++j) s += b[j] * L[j * HD + col];
    bL[col] = s;
  }
}

// ---------------- edge scatter: xagg[dst] += enorm * x[src] (32 f/node) ----------------
__global__ void edge_kernel(const int* __restrict__ src, const int* __restrict__ dst,
                            const float* __restrict__ dinv, const float* __restrict__ x,
                            float* __restrict__ xagg, int E) {
  int tid = blockIdx.x * blockDim.x + threadIdx.x;
  int e = tid >> 2;
  if (e >= E) return;
  int c = (tid & 3) * 8;
  int s = src[e], d = dst[e];
  float w = dinv[s] * dinv[d];
  const float4* xs = (const float4*)(x + (size_t)s * 32 + c);
  float* xd = xagg + (size_t)d * 32 + c;
  float4 a = xs[0];
  float4 b = xs[1];
  atomicAdd(xd + 0, w * a.x);
  atomicAdd(xd + 1, w * a.y);
  atomicAdd(xd + 2, w * a.z);
  atomicAdd(xd + 3, w * a.w);
  atomicAdd(xd + 4, w * b.x);
  atomicAdd(xd + 5, w * b.y);
  atomicAdd(xd + 6, w * b.z);
  atomicAdd(xd + 7, w * b.w);
}

// ---------------- per-node dense A3TGCN with fused gate weights (WMMA) ----------------
__global__ __launch_bounds__(128) void node_kernel(
    const float* __restrict__ x, const float* __restrict__ xagg,
    const float* __restrict__ dinv,
    const float* __restrict__ WzL, const float* __restrict__ bzL,
    const float* __restrict__ WhL, const float* __restrict__ bhL,
    const float* __restrict__ att,
    const float* __restrict__ outW, const float* __restrict__ outb,
    float* __restrict__ out, int N) {
  __shared__ float s_xa[4][512];  // per-wave: 16 nodes x 32 (F_IN x PERIODS); reused for relu(acc)

  const int lane = threadIdx.x & 31;
  const int wave = threadIdx.x >> 5;
  const int tile0 = (blockIdx.x * 4 + wave) * 16;
  float* xa = s_xa[wave];

  // softmax(att) -> attention probs (uniform across lanes)
  float pr[PD];
  {
    float m = att[0];
#pragma unroll
    for (int t = 1; t < PD; ++t) m = fmaxf(m, att[t]);
    float ssum = 0.0f;
#pragma unroll
    for (int t = 0; t < PD; ++t) { pr[t] = expf(att[t] - m); ssum += pr[t]; }
    float inv = 1.0f / ssum;
#pragma unroll
    for (int t = 0; t < PD; ++t) pr[t] *= inv;
  }

  const int c0 = lane & 15;        // column (B/C) or M-row (A) owned by this lane
  const int kb = (lane >> 4) * 2;  // K pair base: lanes 0-15 -> {0,1}, 16-31 -> {2,3}

  // ---- preload fused B fragments (4x32 each) + per-column biases ----
  v2f wzf[2], whf[2];
  float bzv[2], bhv[2];
#pragma unroll
  for (int n = 0; n < 2; ++n) {
    int col = n * 16 + c0;
    wzf[n].x = WzL[(kb + 0) * HD + col];
    wzf[n].y = WzL[(kb + 1) * HD + col];
    whf[n].x = WhL[(kb + 0) * HD + col];
    whf[n].y = WhL[(kb + 1) * HD + col];
    bzv[n] = bzL[col];
    bhv[n] = bhL[col];
  }

  // ---- stage xa = xagg + snorm * x for this wave's 16 nodes ----
#pragma unroll
  for (int i = 0; i < 4; ++i) {
    int idx = lane + i * 32;  // float4 chunk id 0..127
    int nl = idx >> 3;
    int ch = idx & 7;
    int node = tile0 + nl;
    int nc = node < N ? node : (N - 1);
    float dv = dinv[nc];
    float sn = dv * dv;
    float4 xv = ((const float4*)(x + (size_t)nc * 32))[ch];
    float4 av = ((const float4*)(xagg + (size_t)nc * 32))[ch];
    float4 r;
    r.x = av.x + sn * xv.x;
    r.y = av.y + sn * xv.y;
    r.z = av.z + sn * xv.z;
    r.w = av.w + sn * xv.w;
    ((float4*)(xa + nl * 32))[ch] = r;
  }
  __syncthreads();

  const v8f czero = splat8(0.0f);  // one shared zero C operand for all WMMAs
  v8f acc0 = czero, acc1 = czero;

#pragma unroll
  for (int t = 0; t < PD; ++t) {
    // A fragment (16x4 f32): lane holds row c0, K pair {kb, kb+1}; feat stride = PD
    v2f a;
    a.x = xa[c0 * 32 + (kb + 0) * PD + t];
    a.y = xa[c0 * 32 + (kb + 1) * PD + t];

    // u = xa_t @ WzL ; v = xa_t @ WhL   (bias folded into elementwise below)
    v8f u0 = wmma4(a, wzf[0], czero);
    v8f u1 = wmma4(a, wzf[1], czero);
    v8f v0 = wmma4(a, whf[0], czero);
    v8f v1 = wmma4(a, whf[1], czero);

    // H_t = (1 - sigmoid(u+bz)) * tanh(v+bh);  acc += p_t * H_t   (H0 == 0)
    // branch-free: 1-sigmoid(s) = e^{-s}/(1+e^{-s});  tanh(s) = (e^{2s}-1)/(e^{2s}+1)
#pragma unroll
    for (int r = 0; r < 8; ++r) {
      float eu0 = exp_clamped(-(u0[r] + bzv[0]));
      float eu1 = exp_clamped(-(u1[r] + bzv[1]));
      float omz0 = eu0 * rcp1(1.0f + eu0);
      float omz1 = eu1 * rcp1(1.0f + eu1);
      float tv0 = exp_clamped(2.0f * (v0[r] + bhv[0]));
      float tv1 = exp_clamped(2.0f * (v1[r] + bhv[1]));
      float th0 = (tv0 - 1.0f) * rcp1(tv0 + 1.0f);
      float th1 = (tv1 - 1.0f) * rcp1(tv1 + 1.0f);
      acc0[r] += pr[t] * omz0 * th0;
      acc1[r] += pr[t] * omz1 * th1;
    }
  }

  // ---- final: out = relu(acc) @ outW + outb ----
  __syncthreads();  // xa reads done; reuse buffer for relu(acc) [16][32]
#pragma unroll
  for (int r = 0; r < 8; ++r) {
    int row = r + ((lane >> 4) << 3);
    xa[row * 32 + c0] = fmaxf(acc0[r], 0.0f);
    xa[row * 32 + 16 + c0] = fmaxf(acc1[r], 0.0f);
  }
  __syncthreads();

  {
    int nl = lane & 15;
    int ph = lane >> 4;  // two lanes per node, 4 output periods each
    int p0 = ph * 4;
    float4 o;
    o.x = outb[p0 + 0];
    o.y = outb[p0 + 1];
    o.z = outb[p0 + 2];
    o.w = outb[p0 + 3];
#pragma unroll
    for (int k = 0; k < HD; ++k) {
      float aL = xa[nl * 32 + k];
      float4 w = *(const float4*)(outW + k * PD + p0);
      o.x += aL * w.x;
      o.y += aL * w.y;
      o.z += aL * w.z;
      o.w += aL * w.w;
    }
    int node = tile0 + nl;
    if (node < N) *((float4*)(out + (size_t)node * PD + p0)) = o;
  }
}

extern "C" void kernel_launch(void* const* d_in, const int* in_sizes, int n_in,
                              void* d_out, int out_size, void* d_ws, size_t ws_size,
                              hipStream_t stream) {
  const float* x = (const float*)d_in[0];
  const int* ei = (const int*)d_in[1];
  const float* Wz = (const float*)d_in[2];
  const float* bzp = (const float*)d_in[3];
  // d_in[4], d_in[5] (Wr, br) and d_in[10], d_in[11] (LrW, Lrb) are dead: H0==0
  const float* Wh = (const float*)d_in[6];
  const float* bhp = (const float*)d_in[7];
  const float* LzW = (const float*)d_in[8];
  const float* Lzb = (const float*)d_in[9];
  const float* LhW = (const float*)d_in[12];
  const float* Lhb = (const float*)d_in[13];
  const float* att = (const float*)d_in[14];
  const float* outW = (const float*)d_in[15];
  const float* outb = (const float*)d_in[16];
  float* out = (float*)d_out;

  const int N = in_sizes[0] / 32;  // [N, 4, 8]
  const int E = in_sizes[1] / 2;   // [2, E]
  const int* src = ei;
  const int* dst = ei + E;

  // workspace layout: [deg N][xagg 32N][dinv N][WzL 128][bzL 32][WhL 128][bhL 32]
  float* w = (float*)d_ws;
  float* deg = w;
  float* xagg = w + (size_t)N;
  float* dinv = w + (size_t)33 * N;
  float* WzL = w + (size_t)34 * N;
  float* bzL = WzL + 128;
  float* WhL = bzL + 32;
  float* bhL = WhL + 128;

  long long nzero = (long long)33 * N;  // deg + xagg contiguous
  zero_kernel<<<1024, 256, 0, stream>>>(w, nzero);

  fuse_kernel<<<1, 320, 0, stream>>>(Wz, bzp, LzW, Lzb, Wh, bhp, LhW, Lhb,
                                     WzL, bzL, WhL, bhL);

  deg_kernel<<<(E + 255) / 256, 256, 0, stream>>>(dst, deg, E);
  dinv_kernel<<<(N + 255) / 256, 256, 0, stream>>>(deg, dinv, N);

  int eth = E * 4;  // 4 threads per edge, 8 floats each
  edge_kernel<<<(eth + 255) / 256, 256, 0, stream>>>(src, dst, dinv, x, xagg, E);

  int nblocks = (N + 63) / 64;  // 4 waves x 16 nodes per block
  node_kernel<<<nblocks, 128, 0, stream>>>(x, xagg, dinv, WzL, bzL, WhL, bhL,
                                           att, outW, outb, out, N);
}